// ASAPooling_88656714924908
// MI455X (gfx1250) — compile-verified
//
#include <hip/hip_runtime.h>
#include <math.h>

// ---- problem constants from the reference ----
constexpr int N   = 4096;
constexpr int D   = 128;
constexpr int DEG = 12;
constexpr int E0  = N * DEG;   // 49152 input edges
constexpr int E   = E0 + N;    // + N self loops
constexpr int K   = N / 2;     // ceil(0.5*N) = 2048
constexpr float NEG_SLOPE = 0.2f;

typedef float v2f __attribute__((ext_vector_type(2)));
typedef float v8f __attribute__((ext_vector_type(8)));

// ---------------- helpers ----------------
__device__ inline float wave_sum(float v) {
#pragma unroll
  for (int off = 16; off > 0; off >>= 1) v += __shfl_xor(v, off, 32);
  return v;
}
__device__ inline int wave_sum_i(int v) {
#pragma unroll
  for (int off = 16; off > 0; off >>= 1) v += __shfl_xor(v, off, 32);
  return v;
}
__device__ inline void edge_rc(int e, const int* __restrict__ row,
                               const int* __restrict__ col, int& r, int& c) {
  if (e < E0) { r = row[e]; c = col[e]; }
  else        { r = e - E0; c = e - E0; }   // appended self loops
}
// float atomic max via monotone int mapping; init bit pattern 0xFFFFFFFF
__device__ inline void atomicMaxF(float* addr, float val) {
  if (val >= 0.0f) atomicMax((int*)addr, __float_as_int(val));
  else             atomicMin((unsigned int*)addr, __float_as_uint(val));
}

// ---------------- kernels ----------------
// x_q = segment_max(x[row], col)
__global__ void k_seg_max(const float* __restrict__ x, const int* __restrict__ row,
                          const int* __restrict__ col, float* __restrict__ xq) {
  int idx = blockIdx.x * blockDim.x + threadIdx.x;
  if (idx >= E * D) return;
  int e = idx / D, j = idx - e * D;
  int r, c; edge_rc(e, row, col, r, c);
  atomicMaxF(&xq[c * D + j], x[r * D + j]);
}

// Out = A[N,D] @ W[D,D] + bias  -- f32 WMMA, one wave per 16x16 tile
__global__ void k_gemm1_wmma(const float* __restrict__ A, const float* __restrict__ W,
                             const float* __restrict__ bias, float* __restrict__ Out) {
  int wid   = blockIdx.x * (blockDim.x >> 5) + (threadIdx.x >> 5);
  int lane  = threadIdx.x & 31;
  int mtile = wid >> 3, ntile = wid & 7;      // 256 x 8 tiles
  int m0 = mtile * 16, n0 = ntile * 16;
  int hl = lane >> 4, l16 = lane & 15, koff = hl * 2;
  v8f acc = {};
#pragma unroll 4
  for (int k0 = 0; k0 < D; k0 += 4) {
    v2f a = *(const v2f*)(A + (m0 + l16) * D + k0 + koff);  // A rows, K pair
    v2f b;
    b.x = W[(k0 + koff + 0) * D + n0 + l16];                // B rows striped on lanes
    b.y = W[(k0 + koff + 1) * D + n0 + l16];
    acc = __builtin_amdgcn_wmma_f32_16x16x4_f32(false, a, false, b, (short)0, acc,
                                                false, false);
  }
  int gn = n0 + l16;
  float bb = bias[gn];
#pragma unroll
  for (int v = 0; v < 8; ++v) Out[(m0 + v + 8 * hl) * D + gn] = acc[v] + bb;
}

// p[n] = xql[n]·w_att[:D], q[n] = x[n]·w_att[D:]
__global__ void k_node_dots(const float* __restrict__ xql, const float* __restrict__ x,
                            const float* __restrict__ watt, float* pN, float* qN) {
  int n = blockIdx.x * (blockDim.x >> 5) + (threadIdx.x >> 5);
  int lane = threadIdx.x & 31;
  if (n >= N) return;
  float sp = 0.f, sq = 0.f;
  for (int j = lane; j < D; j += 32) {
    sp += xql[n * D + j] * watt[j];
    sq += x[n * D + j] * watt[D + j];
  }
  sp = wave_sum(sp); sq = wave_sum(sq);
  if (lane == 0) { pN[n] = sp; qN[n] = sq; }
}

__global__ void k_edge_score(const int* row, const int* col, const float* pN,
                             const float* qN, const float* batt,
                             float* sc, float* mx) {
  int e = blockIdx.x * blockDim.x + threadIdx.x;
  if (e >= E) return;
  int r, c; edge_rc(e, row, col, r, c);
  float s = pN[c] + qN[r] + batt[0];
  s = (s > 0.f) ? s : NEG_SLOPE * s;          // leaky_relu
  sc[e] = s;
  atomicMaxF(&mx[c], s);
}

__global__ void k_edge_exp(const int* row, const int* col, const float* mx,
                           float* sc, float* ssum) {
  int e = blockIdx.x * blockDim.x + threadIdx.x;
  if (e >= E) return;
  int r, c; edge_rc(e, row, col, r, c);
  float v = expf(sc[e] - mx[c]);
  sc[e] = v;
  atomicAdd(&ssum[c], v);
}

__global__ void k_edge_norm(const int* row, const int* col, const float* ssum,
                            float* sc) {
  int e = blockIdx.x * blockDim.x + threadIdx.x;
  if (e >= E) return;
  int r, c; edge_rc(e, row, col, r, c);
  sc[e] = sc[e] / ssum[c];
}

// x_new = segment_sum(x[row] * score, col)
__global__ void k_xnew(const float* __restrict__ x, const int* row, const int* col,
                       const float* sc, float* __restrict__ xn) {
  int idx = blockIdx.x * blockDim.x + threadIdx.x;
  if (idx >= E * D) return;
  int e = idx / D, j = idx - e * D;
  int r, c; edge_rc(e, row, col, r, c);
  atomicAdd(&xn[c * D + j], x[r * D + j] * sc[e]);
}

__global__ void k_leconv_dots(const float* __restrict__ xn, const float* W1,
                              const float* b1, const float* W2, const float* W3,
                              const float* b3, float* av, float* bv, float* w3v) {
  int n = blockIdx.x * (blockDim.x >> 5) + (threadIdx.x >> 5);
  int lane = threadIdx.x & 31;
  if (n >= N) return;
  float s1 = 0.f, s2 = 0.f, s3 = 0.f;
  for (int j = lane; j < D; j += 32) {
    float v = xn[n * D + j];
    s1 += v * W1[j]; s2 += v * W2[j]; s3 += v * W3[j];
  }
  s1 = wave_sum(s1); s2 = wave_sum(s2); s3 = wave_sum(s3);
  if (lane == 0) { av[n] = s1 + b1[0]; bv[n] = s2; w3v[n] = s3 + b3[0]; }
}

__global__ void k_leconv_agg(const int* row, const int* col, const float* av,
                             const float* bv, float* agg) {
  int e = blockIdx.x * blockDim.x + threadIdx.x;
  if (e >= E) return;
  int r, c; edge_rc(e, row, col, r, c);
  atomicAdd(&agg[c], av[r] - bv[c]);
}

__global__ void k_fitness(const float* agg, const float* w3v, float* fit) {
  int n = blockIdx.x * blockDim.x + threadIdx.x;
  if (n >= N) return;
  float v = agg[n] + w3v[n];
  fit[n] = 1.0f / (1.0f + expf(-v));
}

// exact top-k via rank counting, jax tie-break (lower index first)
__global__ void k_topk(const float* __restrict__ fit, int* perm, float* fitk) {
  int i = blockIdx.x * (blockDim.x >> 5) + (threadIdx.x >> 5);
  int lane = threadIdx.x & 31;
  if (i >= N) return;
  float fi = fit[i];
  int cnt = 0;
  for (int j = lane; j < N; j += 32) {
    float fj = fit[j];
    cnt += (fj > fi) || (fj == fi && j < i);
  }
  cnt = wave_sum_i(cnt);
  if (lane == 0 && cnt < K) { perm[cnt] = i; fitk[cnt] = fi; }
}

__global__ void k_mask(const int* perm, int* mask) {
  int r = blockIdx.x * blockDim.x + threadIdx.x;
  if (r >= K) return;
  mask[perm[r]] = r;
}

__global__ void k_xout(const float* __restrict__ xn, const int* perm,
                       const float* fitk, float* __restrict__ xout) {
  int idx = blockIdx.x * blockDim.x + threadIdx.x;
  if (idx >= K * D) return;
  int r = idx / D, j = idx - r * D;
  xout[idx] = xn[perm[r] * D + j] * fitk[r];
}

// dense S[N,K]: S[row, mask[col]] += score  (invalid -> +0 at column 0, as reference)
__global__ void k_build_S(const int* row, const int* col, const float* sc,
                          const int* mask, float* S) {
  int e = blockIdx.x * blockDim.x + threadIdx.x;
  if (e >= E) return;
  int r, c; edge_rc(e, row, col, r, c);
  int cm = mask[c];
  float s = (cm >= 0) ? sc[e] : 0.0f;
  atomicAdd(&S[(size_t)r * K + (cm >= 0 ? cm : 0)], s);
}

// AS = A @ S via edge list: AS[r,:] += ew * S[c,:]
__global__ void k_AS(const int* row, const int* col, const float* ew,
                     const float* __restrict__ S, float* __restrict__ AS) {
  int e = blockIdx.x;
  int r, c; edge_rc(e, row, col, r, c);
  float w = (e < E0) ? ew[e] : 1.0f;
  const float* Sc = S + (size_t)c * K;
  float* ASr = AS + (size_t)r * K;
  for (int j = threadIdx.x; j < K; j += blockDim.x) atomicAdd(&ASr[j], w * Sc[j]);
}

// A_new = S^T @ AS  (2048 x 2048 x 4096 dense f32 WMMA).
// 2x2 register-blocked: each wave produces a 32x32 output block (4 tiles),
// reusing each A/B fragment twice -> 8 FLOP/byte out of L2. Diagonal zeroed.
__global__ void k_gemm2_wmma(const float* __restrict__ S, const float* __restrict__ AS,
                             float* __restrict__ Anew) {
  int wid   = blockIdx.x * (blockDim.x >> 5) + (threadIdx.x >> 5);  // 0..4095
  int lane  = threadIdx.x & 31;
  int mblk = wid >> 6, nblk = wid & 63;       // 64 x 64 blocks of 32x32
  int m0 = mblk * 32, n0 = nblk * 32;
  int hl = lane >> 4, l16 = lane & 15, koff = hl * 2;
  v8f acc00 = {}, acc01 = {}, acc10 = {}, acc11 = {};
#pragma unroll 2
  for (int r0 = 0; r0 < N; r0 += 4) {
    int rA = r0 + koff;
    const float* S0 = S  + (size_t)rA * K;    // row rA   of S  (A fragment source)
    const float* S1 = S0 + K;                 // row rA+1
    const float* B0 = AS + (size_t)rA * K;
    const float* B1 = B0 + K;
    // prefetch two k-steps ahead (gfx1250 global_prefetch_b8 path)
    __builtin_prefetch(S0 + 8 * (size_t)K + m0 + l16, 0, 3);
    __builtin_prefetch(B0 + 8 * (size_t)K + n0 + l16, 0, 3);
    v2f a0, a1, b0, b1;
    a0.x = S0[m0 + l16];        a0.y = S1[m0 + l16];        // A tile m
    a1.x = S0[m0 + 16 + l16];   a1.y = S1[m0 + 16 + l16];   // A tile m+16
    b0.x = B0[n0 + l16];        b0.y = B1[n0 + l16];        // B tile n
    b1.x = B0[n0 + 16 + l16];   b1.y = B1[n0 + 16 + l16];   // B tile n+16
    acc00 = __builtin_amdgcn_wmma_f32_16x16x4_f32(false, a0, false, b0, (short)0,
                                                  acc00, false, false);
    acc01 = __builtin_amdgcn_wmma_f32_16x16x4_f32(false, a0, false, b1, (short)0,
                                                  acc01, false, false);
    acc10 = __builtin_amdgcn_wmma_f32_16x16x4_f32(false, a1, false, b0, (short)0,
                                                  acc10, false, false);
    acc11 = __builtin_amdgcn_wmma_f32_16x16x4_f32(false, a1, false, b1, (short)0,
                                                  acc11, false, false);
  }
  int gnA = n0 + l16, gnB = n0 + 16 + l16;
#pragma unroll
  for (int v = 0; v < 8; ++v) {
    int gmA = m0 + v + 8 * hl;
    int gmB = gmA + 16;
    Anew[(size_t)gmA * K + gnA] = (gmA == gnA) ? 0.0f : acc00[v];
    Anew[(size_t)gmA * K + gnB] = (gmA == gnB) ? 0.0f : acc01[v];
    Anew[(size_t)gmB * K + gnA] = (gmB == gnA) ? 0.0f : acc10[v];
    Anew[(size_t)gmB * K + gnB] = (gmB == gnB) ? 0.0f : acc11[v];
  }
}

__global__ void k_perm_out(const int* perm, float* out) {
  int r = blockIdx.x * blockDim.x + threadIdx.x;
  if (r >= K) return;
  out[r] = (float)perm[r];
}

// ---------------- host ----------------
extern "C" void kernel_launch(void* const* d_in, const int* in_sizes, int n_in,
                              void* d_out, int out_size, void* d_ws, size_t ws_size,
                              hipStream_t stream) {
  (void)in_sizes; (void)n_in; (void)out_size; (void)ws_size;
  const float* x    = (const float*)d_in[0];
  const int*   ei   = (const int*)d_in[1];
  const float* ew   = (const float*)d_in[2];
  const float* Wlin = (const float*)d_in[3];
  const float* blin = (const float*)d_in[4];
  const float* watt = (const float*)d_in[5];
  const float* batt = (const float*)d_in[6];
  const float* W1   = (const float*)d_in[7];
  const float* b1   = (const float*)d_in[8];
  const float* W2   = (const float*)d_in[9];
  const float* W3   = (const float*)d_in[10];
  const float* b3   = (const float*)d_in[11];
  const int* row = ei;
  const int* col = ei + E0;

  char* wptr = (char*)d_ws;
  auto alloc = [&](size_t bytes) {
    char* r = wptr; wptr += (bytes + 255) & ~(size_t)255; return (void*)r;
  };
  float* xq   = (float*)alloc((size_t)N * D * 4);
  float* xql  = (float*)alloc((size_t)N * D * 4);
  float* xn   = (float*)alloc((size_t)N * D * 4);
  float* pN   = (float*)alloc(N * 4);
  float* qN   = (float*)alloc(N * 4);
  float* mx   = (float*)alloc(N * 4);
  float* ssum = (float*)alloc(N * 4);
  float* sc   = (float*)alloc((size_t)E * 4);
  float* av   = (float*)alloc(N * 4);
  float* bv   = (float*)alloc(N * 4);
  float* w3v  = (float*)alloc(N * 4);
  float* agg  = (float*)alloc(N * 4);
  float* fit  = (float*)alloc(N * 4);
  int*   perm = (int*)alloc(K * 4);
  float* fitk = (float*)alloc(K * 4);
  int*   mask = (int*)alloc(N * 4);
  float* S    = (float*)alloc((size_t)N * K * 4);   // 32 MB (L2-resident)
  float* AS   = (float*)alloc((size_t)N * K * 4);   // 32 MB (L2-resident)

  float* x_out = (float*)d_out;                 // [K, D]
  float* Anew  = x_out + (size_t)K * D;         // [K, K]
  float* permf = Anew + (size_t)K * K;          // [K]

  const int TB = 256;
  // segment max into x_q (0xFF fill is valid init for int-mapped float max)
  hipMemsetAsync(xq, 0xFF, (size_t)N * D * 4, stream);
  k_seg_max<<<(E * D + TB - 1) / TB, TB, 0, stream>>>(x, row, col, xq);
  // x_q @ W_lin + b  (f32 WMMA, 2048 tiles exactly -> EXEC all ones)
  k_gemm1_wmma<<<(N / 16) * (D / 16) / 8, TB, 0, stream>>>(xq, Wlin, blin, xql);
  // attention scores + segment softmax
  k_node_dots<<<N / 8, TB, 0, stream>>>(xql, x, watt, pN, qN);
  hipMemsetAsync(mx, 0xFF, N * 4, stream);
  k_edge_score<<<(E + TB - 1) / TB, TB, 0, stream>>>(row, col, pN, qN, batt, sc, mx);
  hipMemsetAsync(ssum, 0, N * 4, stream);
  k_edge_exp<<<(E + TB - 1) / TB, TB, 0, stream>>>(row, col, mx, sc, ssum);
  k_edge_norm<<<(E + TB - 1) / TB, TB, 0, stream>>>(row, col, ssum, sc);
  // x_new
  hipMemsetAsync(xn, 0, (size_t)N * D * 4, stream);
  k_xnew<<<(E * D + TB - 1) / TB, TB, 0, stream>>>(x, row, col, sc, xn);
  // LEConv fitness
  k_leconv_dots<<<N / 8, TB, 0, stream>>>(xn, W1, b1, W2, W3, b3, av, bv, w3v);
  hipMemsetAsync(agg, 0, N * 4, stream);
  k_leconv_agg<<<(E + TB - 1) / TB, TB, 0, stream>>>(row, col, av, bv, agg);
  k_fitness<<<(N + TB - 1) / TB, TB, 0, stream>>>(agg, w3v, fit);
  // top-k + mask
  k_topk<<<N / 8, TB, 0, stream>>>(fit, perm, fitk);
  hipMemsetAsync(mask, 0xFF, N * 4, stream);
  k_mask<<<(K + TB - 1) / TB, TB, 0, stream>>>(perm, mask);
  // x_out
  k_xout<<<(K * D + TB - 1) / TB, TB, 0, stream>>>(xn, perm, fitk, x_out);
  // S, AS = A@S, A_new = S^T @ AS (dense WMMA GEMM), perm
  hipMemsetAsync(S, 0, (size_t)N * K * 4, stream);
  k_build_S<<<(E + TB - 1) / TB, TB, 0, stream>>>(row, col, sc, mask, S);
  hipMemsetAsync(AS, 0, (size_t)N * K * 4, stream);
  k_AS<<<E, TB, 0, stream>>>(row, col, ew, S, AS);
  k_gemm2_wmma<<<(K / 32) * (K / 32) / 8, TB, 0, stream>>>(S, AS, Anew);
  k_perm_out<<<(K + TB - 1) / TB, TB, 0, stream>>>(perm, permf);
}